// Hamiltonian_87093346828725
// MI455X (gfx1250) — compile-verified
//
#include <hip/hip_runtime.h>

typedef __attribute__((ext_vector_type(2))) float v2f;
typedef __attribute__((ext_vector_type(8))) float v8f;

#define THREADS 256
#define GRID1   1024

// Find row i with off(i) <= p < off(i+1), where off(i) = i*(2n-1-i)/2.
// All quantities fit in u32 for n = 8192 (npairs < 2^26).
// Fast v_sqrt_f32 estimate + exact integer fix-up loops (bit-correct).
__device__ __forceinline__ void findRow32(unsigned int p, int n,
                                          int& io, unsigned int& rso, unsigned int& neo) {
    unsigned int m    = (unsigned int)(2 * n - 1);
    unsigned int disc = m * m - (p << 3);            // exact in u32
    float s = sqrtf((float)disc);
    int i = (int)(((float)m - s) * 0.5f);
    if (i < 0) i = 0;
    if (i > n - 2) i = n - 2;
    unsigned int rs = ((unsigned int)i * (m - (unsigned int)i)) >> 1;
    while (rs > p) { --i; rs = ((unsigned int)i * (m - (unsigned int)i)) >> 1; }
    unsigned int ne = rs + (unsigned int)(n - 1 - i);
    while (p >= ne) { ++i; rs = ne; ne += (unsigned int)(n - 1 - i); }
    io = i; rso = rs; neo = ne;
}

// Reduce the 32 per-lane accumulators of each wave with one V_WMMA_F32_16X16X4_F32.
// A (f32 16x4 layout): lane L<16 holds A[L][0..1], lane L>=16 holds A[L-16][2..3].
// acc goes in the lane's first A dword, second = 0; B = all-ones 4x16
// => D[m][n] = acc(lane m) + acc(lane m+16) for every column n.
// Sum of the 8 D dwords: lanes 0..15 = rows 0..7 total, lanes 16..31 = rows 8..15 total.
__device__ __forceinline__ void waveReduceWMMA(float acc, float* red16) {
    v2f a; a.x = acc;  a.y = 0.0f;
    v2f b; b.x = 1.0f; b.y = 1.0f;
    v8f c = {};
    c = __builtin_amdgcn_wmma_f32_16x16x4_f32(false, a, false, b, (short)0, c, false, false);
    float s = c[0] + c[1] + c[2] + c[3] + c[4] + c[5] + c[6] + c[7];
    int lane = threadIdx.x & 31;
    int warp = threadIdx.x >> 5;
    if (lane == 0)  red16[warp * 2 + 0] = s;
    if (lane == 16) red16[warp * 2 + 1] = s;
}

// Kernel 1: stream the C(n,2) pair coefficients (134 MB) at HBM rate.
// Each thread: 16 contiguous pair indices per iteration (4x global_load_b128),
// one sqrt row-inversion per 16, then incremental u32 counters. Since p steps
// by 1 per element and every row has length >= 1, at most one row crossing per
// element => a single rare `if` (no inner while).
__global__ void __launch_bounds__(THREADS)
ham_pairs_kernel(const float* __restrict__ x, const float* __restrict__ c2,
                 float* __restrict__ blockOut, int n, int ngrp16)
{
    __shared__ float xl[8192];
    __shared__ float red[16];

    // Stage x into LDS (32 KB) with b128 loads.
    const float4* x4  = (const float4*)x;
    float4*       xl4 = (float4*)xl;
    int nq = n >> 2;
    for (int t = threadIdx.x; t < nq; t += THREADS) xl4[t] = x4[t];
    __syncthreads();

    float acc = 0.0f;
    const unsigned int stride = gridDim.x * THREADS;
    for (unsigned int q = blockIdx.x * THREADS + threadIdx.x; q < (unsigned int)ngrp16; q += stride) {
        unsigned int p0 = q << 4;
        const float4* cp = (const float4*)(c2 + p0);
        if (q + stride < (unsigned int)ngrp16) {
            __builtin_prefetch((const void*)(c2 + ((size_t)(q + stride) << 4)), 0, 1); // global_prefetch_b8
        }
        float4 c0 = cp[0], c1 = cp[1], cc2 = cp[2], c3 = cp[3];
        float cv[16] = {c0.x, c0.y, c0.z, c0.w,  c1.x, c1.y, c1.z, c1.w,
                        cc2.x, cc2.y, cc2.z, cc2.w,  c3.x, c3.y, c3.z, c3.w};

        int i; unsigned int rs, ne;
        findRow32(p0, n, i, rs, ne);
        unsigned int j   = p0 - rs + (unsigned int)(i + 1); // column of element p0
        unsigned int rem = ne - p0;                         // elements left in this row
        float xi = xl[i];

        #pragma unroll
        for (int e = 0; e < 16; ++e) {
            acc = fmaf(cv[e], xi * xl[j], acc);
            --rem;
            ++j;
            if (rem == 0) {                // rare: advance exactly one row
                ++i;
                xi  = xl[i];
                j   = (unsigned int)(i + 1);
                rem = (unsigned int)(n - 1 - i);
            }
        }
    }

    // Loop reconverged: full EXEC for the WMMA reduction.
    waveReduceWMMA(acc, red);
    __syncthreads();
    if (threadIdx.x == 0) {
        float t = 0.0f;
        #pragma unroll
        for (int r = 0; r < 16; ++r) t += red[r];
        blockOut[blockIdx.x] = t;
    }
}

// Kernel 2: degree-1 dot + pair tail (npairs % 16) + block partials -> d_out[0].
__global__ void __launch_bounds__(THREADS)
ham_final_kernel(const float* __restrict__ x, const float* __restrict__ coeffs,
                 const float* __restrict__ blockPart, float* __restrict__ out,
                 int n, int nblocks, unsigned int npairs, unsigned int remStart)
{
    __shared__ float red[16];
    float acc = 0.0f;

    for (int t = threadIdx.x; t < n; t += THREADS)
        acc = fmaf(coeffs[t], x[t], acc);

    for (int t = threadIdx.x; t < nblocks; t += THREADS)
        acc += blockPart[t];

    const float* c2 = coeffs + n;
    for (unsigned int p = remStart + threadIdx.x; p < npairs; p += THREADS) {
        int i; unsigned int rs, ne;
        findRow32(p, n, i, rs, ne);
        int j = (int)(p - rs) + i + 1;
        acc = fmaf(c2[p], x[i] * x[j], acc);
    }

    waveReduceWMMA(acc, red);
    __syncthreads();
    if (threadIdx.x == 0) {
        float t = 0.0f;
        #pragma unroll
        for (int r = 0; r < 16; ++r) t += red[r];
        out[0] = t;
    }
}

extern "C" void kernel_launch(void* const* d_in, const int* in_sizes, int n_in,
                              void* d_out, int out_size, void* d_ws, size_t ws_size,
                              hipStream_t stream) {
    const float* x      = (const float*)d_in[0];
    const float* coeffs = (const float*)d_in[1];
    float*       out    = (float*)d_out;

    int n = in_sizes[0];                                    // 8192
    unsigned int npairs   = (unsigned int)((long long)n * (n - 1) / 2);
    int          ngrp16   = (int)(npairs >> 4);             // full groups of 16
    unsigned int remStart = (unsigned int)ngrp16 << 4;

    int grid = GRID1;
    size_t maxBlocks = ws_size / sizeof(float);
    if (maxBlocks < (size_t)grid) grid = (int)(maxBlocks > 0 ? maxBlocks : 1);

    float* blockPart = (float*)d_ws;

    ham_pairs_kernel<<<grid, THREADS, 0, stream>>>(x, coeffs + n, blockPart, n, ngrp16);
    ham_final_kernel<<<1, THREADS, 0, stream>>>(x, coeffs, blockPart, out,
                                                n, grid, npairs, remStart);
}